// DarkChannelPrior_82446192214524
// MI455X (gfx1250) — compile-verified
//
#include <hip/hip_runtime.h>

// ---------------- problem constants ----------------
#define H     1024
#define W     1024
#define HW    (H*W)
#define PAD   7
#define PATCH 15
#define TS    64              // output tile (64x64)
#define RV    (TS + PATCH - 1) // 78 valid halo rows/cols
#define RR    80              // allocated region rows (pad to 5 x 16 blocks)
#define RS    79              // region stride in floats (odd -> bank-conflict free)
#define NELEM (RR*RS)         // 6320 floats per region
#define KITER ((NELEM + 255)/256) // 25 staging iters per thread

typedef __attribute__((ext_vector_type(2))) float v2f;
typedef __attribute__((ext_vector_type(8))) float v8f;

// ---- feature gates (device pass only; host pass parses fallbacks) ----
#if defined(__HIP_DEVICE_COMPILE__) && __has_builtin(__builtin_amdgcn_global_load_async_to_lds_b32)
#define USE_ASYNC_LDS 1
#endif
#if defined(__HIP_DEVICE_COMPILE__) && __has_builtin(__builtin_amdgcn_wmma_f32_16x16x4_f32)
#define USE_WMMA4 1
#endif

__device__ __forceinline__ int reflect_idx(int c) {
  // jnp.pad(mode="reflect"), pad=7 < H
  if (c < 0)  c = -c;
  if (c >= H) c = 2*H - 2 - c;
  return c;
}

__device__ __forceinline__ void async_g2l_b32(const float* g, float* l) {
#ifdef USE_ASYNC_LDS
  __builtin_amdgcn_global_load_async_to_lds_b32(
      (__attribute__((address_space(1))) int*)(g),
      (__attribute__((address_space(3))) int*)(l),
      0, 0);
#else
  *l = *g;
#endif
}

__device__ __forceinline__ void wait_async_then_barrier() {
#ifdef USE_ASYNC_LDS
#if __has_builtin(__builtin_amdgcn_s_wait_asynccnt)
  __builtin_amdgcn_s_wait_asynccnt(0);
#else
  asm volatile("s_wait_asynccnt 0" ::: "memory");
#endif
#endif
  __syncthreads();
}

__global__ __launch_bounds__(256)
void DarkChannelPrior_kernel(const float* __restrict__ x, float* __restrict__ out) {
  // two regions: [0..NELEM) = ch0 then xmin ; [NELEM..2*NELEM) = ch1 then hsum
  __shared__ float smem[2 * NELEM];     // 50,560 bytes
  float* xmin = smem;
  float* hsum = smem + NELEM;

  const int tid = threadIdx.x;
  const int tx  = blockIdx.x;           // tile col
  const int ty  = blockIdx.y;           // tile row
  const int b   = blockIdx.z;           // batch

  const float* xb = x + (size_t)b * 3 * HW;

  // ---- Phase 1: stage ch0/ch1 into LDS (async), ch2 into registers ----
  float c2v[KITER];
#pragma unroll
  for (int k = 0; k < KITER; ++k) {
    c2v[k] = 0.0f;
    int idx = tid + k * 256;
    if (idx < NELEM) {
      int ry = idx / RS;
      int rx = idx - ry * RS;
      bool valid = (ry < RV) && (rx < RV);
      if (valid) {
        int gy = reflect_idx(ty * TS + ry - PAD);
        int gx = reflect_idx(tx * TS + rx - PAD);
        const float* src = xb + (size_t)gy * W + gx;
        async_g2l_b32(src,        &xmin[idx]);
        async_g2l_b32(src + HW,   &hsum[idx]);   // hsum region holds ch1 for now
        c2v[k] = src[2 * HW];
      } else {
        xmin[idx] = 0.0f;                        // defined zero padding
        hsum[idx] = 0.0f;
      }
    }
  }
  wait_async_then_barrier();

  // ---- Phase 2: 3-way channel min, in place into xmin ----
#pragma unroll
  for (int k = 0; k < KITER; ++k) {
    int idx = tid + k * 256;
    if (idx < NELEM)
      xmin[idx] = fminf(fminf(xmin[idx], hsum[idx]), c2v[k]);
  }
  __syncthreads();                               // hsum region now free for reuse

#ifdef USE_WMMA4
  const int lane = tid & 31;
  const int wv   = tid >> 5;                     // 8 waves
  const int hf   = lane >> 4;                    // half-wave select
  const int mn   = lane & 15;                    // M (A rows) / N (B cols)

  // ---- Phase 3: horizontal 15-sum via V_WMMA_F32_16X16X4_F32 ----
  // hsum[ry][cx] = sum_{j=0..14} xmin[ry][cx+j]; blocks: 5 row x 4 col
  for (int blk = wv; blk < 20; blk += 8) {
    int rb = blk >> 2, cb = blk & 3;
    int rbase = rb * 16, cbase = cb * 16;
    v8f acc = {0.f,0.f,0.f,0.f,0.f,0.f,0.f,0.f};
#pragma unroll
    for (int t = 0; t < 8; ++t) {                // K = 32 in chunks of 4
      int kk = 4 * t + 2 * hf;
      v2f a, bm;
      // A (data): A[m][k] = xmin[rbase+m][cbase+k]; lanes 0-15 hold k,k+1 ; 16-31 hold k+2,k+3
      a.x = xmin[(rbase + mn) * RS + cbase + kk];
      a.y = xmin[(rbase + mn) * RS + cbase + kk + 1];
      // B (constant band): B[k][n] = 1 iff 0 <= k-n <= 14
      int d0 = kk - mn;
      bm.x = (d0 >= 0 && d0 <= 14) ? 1.0f : 0.0f;
      bm.y = (d0 + 1 >= 0 && d0 + 1 <= 14) ? 1.0f : 0.0f;
      acc = __builtin_amdgcn_wmma_f32_16x16x4_f32(false, a, false, bm,
                                                  (short)0, acc, false, false);
    }
#pragma unroll
    for (int v = 0; v < 8; ++v)
      hsum[(rbase + v + hf * 8) * RS + cbase + mn] = acc[v];
  }
  __syncthreads();

  // ---- Phase 4: vertical 15-sum via WMMA, store to global ----
  // dark[r][c] = sum_{j=0..14} hsum[r+j][c]; blocks: 4 row x 4 col
  for (int blk = wv; blk < 16; blk += 8) {
    int vb = blk >> 2, cb = blk & 3;
    int vbase = vb * 16, cbase = cb * 16;
    v8f acc = {0.f,0.f,0.f,0.f,0.f,0.f,0.f,0.f};
#pragma unroll
    for (int t = 0; t < 8; ++t) {
      int kk = 4 * t + 2 * hf;
      v2f a, bm;
      // A (constant band): A[m][k] = 1 iff 0 <= k-m <= 14
      int d0 = kk - mn;
      a.x = (d0 >= 0 && d0 <= 14) ? 1.0f : 0.0f;
      a.y = (d0 + 1 >= 0 && d0 + 1 <= 14) ? 1.0f : 0.0f;
      // B (data): B[k][n] = hsum[vbase+k][cbase+n]
      bm.x = hsum[(vbase + kk) * RS + cbase + mn];
      bm.y = hsum[(vbase + kk + 1) * RS + cbase + mn];
      acc = __builtin_amdgcn_wmma_f32_16x16x4_f32(false, a, false, bm,
                                                  (short)0, acc, false, false);
    }
#pragma unroll
    for (int v = 0; v < 8; ++v) {
      int oy = ty * TS + vbase + v + hf * 8;     // C/D layout: M = v (+8 for upper half)
      int ox = tx * TS + cbase + mn;             // N = lane&15
      out[((size_t)b * H + oy) * W + ox] = acc[v];
    }
  }
#else
  // ---- scalar separable fallback (also parsed on host pass) ----
  for (int idx = tid; idx < RV * TS; idx += 256) {
    int ry = idx / TS, cx = idx - ry * TS;
    float s = 0.0f;
#pragma unroll
    for (int j = 0; j < PATCH; ++j) s += xmin[ry * RS + cx + j];
    hsum[ry * RS + cx] = s;
  }
  __syncthreads();
  for (int idx = tid; idx < TS * TS; idx += 256) {
    int r = idx / TS, cx = idx - r * TS;
    float s = 0.0f;
#pragma unroll
    for (int j = 0; j < PATCH; ++j) s += hsum[(r + j) * RS + cx];
    int oy = ty * TS + r, ox = tx * TS + cx;
    out[((size_t)b * H + oy) * W + ox] = s;
  }
#endif
}

extern "C" void kernel_launch(void* const* d_in, const int* in_sizes, int n_in,
                              void* d_out, int out_size, void* d_ws, size_t ws_size,
                              hipStream_t stream) {
  (void)n_in; (void)out_size; (void)d_ws; (void)ws_size;
  const float* x = (const float*)d_in[0];
  float* out = (float*)d_out;
  int B = in_sizes[0] / (3 * HW);                // = 16
  dim3 grid(W / TS, H / TS, B);
  DarkChannelPrior_kernel<<<grid, dim3(256), 0, stream>>>(x, out);
}